// GtTransform_43903155700316
// MI455X (gfx1250) — compile-verified
//
#include <hip/hip_runtime.h>
#include <math.h>

#define NB    16
#define NGT   64
#define NC    80
#define NA    8525
#define KTOP  45
#define NBINS 17
#define IMGF  640.0f
#define A_TILES ((NA + 15) / 16)   /* 533 */

typedef __attribute__((ext_vector_type(2))) float        v2f;
typedef __attribute__((ext_vector_type(8))) float        v8f;
typedef __attribute__((ext_vector_type(4))) unsigned int v4u;
typedef __attribute__((ext_vector_type(8))) int          v8i;
typedef __attribute__((ext_vector_type(4))) int          v4i;

#if __has_builtin(__builtin_amdgcn_tensor_load_to_lds) && __has_builtin(__builtin_amdgcn_s_wait_tensorcnt)
#define USE_TDM 1
#else
#define USE_TDM 0
#endif

__device__ __forceinline__ float iou_f(const float b1[4], const float b2[4]) {
    float ty = fmaxf(b1[0], b2[0]);
    float tx = fmaxf(b1[1], b2[1]);
    float by = fminf(b1[2], b2[2]);
    float bx = fminf(b1[3], b2[3]);
    float inter = fmaxf(by - ty, 0.0f) * fmaxf(bx - tx, 0.0f);
    float a1 = (b1[2] - b1[0]) * (b1[3] - b1[1]);
    float a2 = (b2[2] - b2[0]) * (b2[3] - b2[1]);
    return inter / (a1 + a2 - inter);
}

// ---------------------------------------------------------------- anchors ---
__global__ void k_setup_anchors(float* __restrict__ pts, float* __restrict__ boxes,
                                float* __restrict__ strides,
                                float* __restrict__ out_pts, float* __restrict__ out_strides) {
    int a = blockIdx.x * blockDim.x + threadIdx.x;
    if (a >= NA) return;
    int s, base, w;
    if (a < 6400)      { s = 8;   base = 0;    w = 80; }
    else if (a < 8000) { s = 16;  base = 6400; w = 40; }
    else if (a < 8400) { s = 32;  base = 8000; w = 20; }
    else if (a < 8500) { s = 64;  base = 8400; w = 10; }
    else               { s = 128; base = 8500; w = 5;  }
    int r = a - base;
    int y = r / w, x = r % w;
    float fs = (float)s;
    float py = ((float)y + 0.5f) * fs, px = ((float)x + 0.5f) * fs;
    pts[2 * a + 0] = py;  pts[2 * a + 1] = px;
    boxes[4 * a + 0] = (float)y * fs;
    boxes[4 * a + 1] = (float)x * fs;
    boxes[4 * a + 2] = (float)(y + 1) * fs;
    boxes[4 * a + 3] = (float)(x + 1) * fs;
    strides[a] = fs;
    out_pts[2 * a + 0] = py;  out_pts[2 * a + 1] = px;
    out_strides[a] = fs;
}

// ------------------------------------------------- DFL integral (softmax) ---
__global__ void k_pred_d(const float* __restrict__ pred_reg, float* __restrict__ pred_d) {
    long i = (long)blockIdx.x * blockDim.x + threadIdx.x;
    if (i >= (long)NB * 4 * NA) return;
    int a = (int)(i % NA);
    long bc = i / NA;                        // b*4 + channel
    const float* base = pred_reg + bc * (long)NBINS * NA + a;
    __builtin_prefetch(base, 0, 0);          // -> global_prefetch_b8 (gfx1250)
    float m = -__builtin_inff();
#pragma unroll
    for (int k = 0; k < NBINS; ++k) m = fmaxf(m, base[(long)k * NA]);
    float se = 0.0f, sw = 0.0f;
#pragma unroll
    for (int k = 0; k < NBINS; ++k) {
        float e = __expf(base[(long)k * NA] - m);
        se += e; sw += e * (float)k;
    }
    pred_d[i] = sw / se;
}

// -------------------------------------- squared-distance GEMM via WMMA f32 ---
// dist2 = cy^2+cx^2 + py^2+px^2 - 2(cy*py + cx*px)
//       = [cy, cx, cy^2+cx^2, 1] . [-2py, -2px, 1, py^2+px^2]  (rank-4 GEMM)
__global__ __launch_bounds__(256) void k_dist2(const float* __restrict__ gtb,
                                               const float* __restrict__ pts,
                                               float* __restrict__ dist2) {
    int wave = threadIdx.x >> 5;
    int lane = threadIdx.x & 31;
    int tile = blockIdx.x * 8 + wave;
    const int total = NB * (NGT / 16) * A_TILES;
    if (tile >= total) return;               // wave-uniform: EXEC stays full
    int at   = tile % A_TILES;
    int rest = tile / A_TILES;
    int nt   = rest % (NGT / 16);
    int b    = rest / (NGT / 16);
    int h    = lane & 15;
    bool hi  = lane >= 16;

    // A operand (16x4 f32): lanes 0-15 hold K=0,1; lanes 16-31 hold K=2,3; M = lane&15
    int g = nt * 16 + h;
    const float* gb = gtb + ((size_t)b * NGT + g) * 4;
    float cy = 0.5f * (gb[0] + gb[2]);
    float cx = 0.5f * (gb[1] + gb[3]);
    v2f av;
    av.x = hi ? (cy * cy + cx * cx) : cy;    // K=2 : K=0
    av.y = hi ? 1.0f : cx;                   // K=3 : K=1

    // B operand (4x16 f32): same K grouping; N = lane&15
    int col = at * 16 + h;
    int ac = (col < NA) ? col : (NA - 1);
    float py = pts[2 * ac + 0], px = pts[2 * ac + 1];
    v2f bv;
    bv.x = hi ? 1.0f : (-2.0f * py);
    bv.y = hi ? (py * py + px * px) : (-2.0f * px);

    v8f c = {};
#if __has_builtin(__builtin_amdgcn_wmma_f32_16x16x4_f32)
    c = __builtin_amdgcn_wmma_f32_16x16x4_f32(false, av, false, bv, (short)0, c, false, false);
#else
#pragma unroll
    for (int r = 0; r < 8; ++r) {            // scalar fallback, same layout
        int row = r + (hi ? 8 : 0);
        const float* gb2 = gtb + ((size_t)b * NGT + nt * 16 + row) * 4;
        float cy2 = 0.5f * (gb2[0] + gb2[2]);
        float cx2 = 0.5f * (gb2[1] + gb2[3]);
        float dy = cy2 - py, dx = cx2 - px;
        c[r] = dy * dy + dx * dx;
    }
#endif
    // C/D layout: VGPR r -> M = r (lanes 0-15) / r+8 (lanes 16-31), N = lane&15
    if (col < NA) {
#pragma unroll
        for (int r = 0; r < 8; ++r) {
            int row = r + (hi ? 8 : 0);
            dist2[((size_t)b * NGT + nt * 16 + row) * NA + col] = c[r];
        }
    }
}

// ------------------------------------------------------- scatter buffers ----
__global__ void k_init(int* __restrict__ win_d, int* __restrict__ qfl_int) {
    long i = (long)blockIdx.x * blockDim.x + threadIdx.x;
    if (i < (long)NB * NC * NA) qfl_int[i] = -1;
    if (i < (long)NB * NA)      win_d[i]   = -1;
}

// ------------------------- per-gt top-45 + threshold + scatter-max assign ---
__global__ __launch_bounds__(256) void k_assign(const float* __restrict__ dist2,
                                                const float* __restrict__ boxes,
                                                const float* __restrict__ pts,
                                                const float* __restrict__ gtb,
                                                const int* __restrict__ gtc,
                                                int* __restrict__ win_d,
                                                int* __restrict__ qfl_int) {
    __shared__ float sDist[NA];     // 34.1 KB of the 320 KB WGP LDS
    __shared__ float sVal[256];
    __shared__ int   sIdx[256];
    __shared__ int   sK[KTOP];
    __shared__ float sDg[KTOP];
    __shared__ float sTg;

    int tid = threadIdx.x;
    int b = blockIdx.x / NGT;
    int g = blockIdx.x % NGT;
    const float* row = dist2 + ((size_t)b * NGT + g) * NA;

#if USE_TDM
    // Stage the 8525-float distance row into LDS with one Tensor-DMA descriptor.
    // D# group0: {count=1, lds_addr, global_addr[31:0], global_addr[56:32]|type=2}
    // D# group1: data_size=4B, tensor_dim0=NA, tensor_dim1=1, tile=NAx1, stride0=NA
    if (tid < 32) {                                   // one wave issues the DMA
        unsigned       lds_off = (unsigned)(size_t)&sDist[0];   // LDS aperture: addr[31:0]
        unsigned long long ga  = (unsigned long long)(size_t)row;
        v4u g0;
        g0.x = 1u;                                              // count=1, flags=0
        g0.y = lds_off;                                         // lds_addr
        g0.z = (unsigned)(ga & 0xFFFFFFFFu);                    // global_addr[31:0]
        g0.w = (unsigned)((ga >> 32) & 0x1FFFFFFu) | (2u << 30);// global_addr[56:32], type=2
        v8i g1;
        g1[0] = (int)(2u << 16);                 // data_size=4B; no multicast/pad/iterate
        g1[1] = (int)((NA & 0xFFFF) << 16);      // tensor_dim0[15:0]  (bits 63:48)
        g1[2] = (int)((NA >> 16) | (1 << 16));   // tensor_dim0[31:16], tensor_dim1=1
        g1[3] = (int)(NA << 16);                 // tensor_dim1 hi=0, tile_dim0=NA
        g1[4] = 1;                               // tile_dim1=1, tile_dim2=0
        g1[5] = NA;                              // tensor_dim0_stride[31:0]
        g1[6] = 0;                               // stride0[47:32], stride1 lo
        g1[7] = 0;
        v4i z4 = {0, 0, 0, 0};                   // groups 2/3 unused (<=2D tensor)
        v8i z8 = {0, 0, 0, 0, 0, 0, 0, 0};
        __builtin_amdgcn_tensor_load_to_lds(g0, g1, z4, z4, z8, 0);  // 6-arg form
        __builtin_amdgcn_s_wait_tensorcnt(0);    // DMA complete before barrier
    }
    __syncthreads();
#else
    for (int i = tid; i < NA; i += 256) sDist[i] = row[i];
    __syncthreads();
#endif

    // iterative argmin x45 (stable ties -> lowest index, matching lax.top_k)
    for (int k = 0; k < KTOP; ++k) {
        float best = __builtin_inff(); int bi = NA;
        for (int i = tid; i < NA; i += 256) {
            float d = sDist[i];
            if (d < best) { best = d; bi = i; }
        }
        sVal[tid] = best; sIdx[tid] = bi;
        __syncthreads();
        for (int s = 128; s > 0; s >>= 1) {
            if (tid < s) {
                float ov = sVal[tid + s]; int oi = sIdx[tid + s];
                if (ov < sVal[tid] || (ov == sVal[tid] && oi < sIdx[tid])) {
                    sVal[tid] = ov; sIdx[tid] = oi;
                }
            }
            __syncthreads();
        }
        if (tid == 0) { int w = sIdx[0]; sK[k] = w; sDist[w] = __builtin_inff(); }
        __syncthreads();
    }

    float gy1 = gtb[((size_t)b * NGT + g) * 4 + 0];
    float gx1 = gtb[((size_t)b * NGT + g) * 4 + 1];
    float gy2 = gtb[((size_t)b * NGT + g) * 4 + 2];
    float gx2 = gtb[((size_t)b * NGT + g) * 4 + 3];
    if (tid < KTOP) {
        int a = sK[tid];
        float b1[4] = {gy1, gx1, gy2, gx2};
        float b2[4] = {boxes[4 * a], boxes[4 * a + 1], boxes[4 * a + 2], boxes[4 * a + 3]};
        sDg[tid] = iou_f(b1, b2);
    }
    __syncthreads();
    if (tid == 0) {
        float s0 = 0.0f, s1 = 0.0f;
        for (int k = 0; k < KTOP; ++k) { float v = sDg[k]; s0 += v; s1 += v * v; }
        float mean = s0 / (float)KTOP;
        float var  = (s1 - s0 * s0 / (float)KTOP) / (float)(KTOP - 1);   // ddof=1
        sTg = mean + sqrtf(fmaxf(var, 0.0f));
    }
    __syncthreads();
    if (tid < KTOP) {
        int a = sK[tid];
        float ay = pts[2 * a], ax = pts[2 * a + 1];
        bool inside = (gy1 <= ay) && (ay <= gy2) && (gx1 <= ax) && (ax <= gx2);
        if (sDg[tid] >= sTg && inside) {
            atomicMax(&win_d[(size_t)b * NA + a], g);                     // last gt wins
            int cls = gtc[b * NGT + g];
            atomicMax(&qfl_int[((size_t)b * NC + cls) * NA + a], g);
        }
    }
}

// ---------------------------------------------------------------- DFL out ---
__global__ void k_dfl(const int* __restrict__ win_d, const float* __restrict__ gtb,
                      const float* __restrict__ pts, const float* __restrict__ strides,
                      float* __restrict__ dfl) {
    long i = (long)blockIdx.x * blockDim.x + threadIdx.x;
    if (i >= (long)NB * NA) return;
    int b = (int)(i / NA), a = (int)(i % NA);
    int w = win_d[i];
    float* o = dfl + (size_t)b * 4 * NA + a;
    if (w < 0) {
        float inf = __builtin_inff();
        o[0] = inf; o[(size_t)NA] = inf; o[2 * (size_t)NA] = inf; o[3 * (size_t)NA] = inf;
    } else {
        const float* wb = gtb + ((size_t)b * NGT + w) * 4;
        float py = pts[2 * a], px = pts[2 * a + 1], s = strides[a];
        o[0]              = (py - wb[0]) / s;
        o[(size_t)NA]     = (px - wb[1]) / s;
        o[2 * (size_t)NA] = (wb[2] - py) / s;
        o[3 * (size_t)NA] = (wb[3] - px) / s;
    }
}

// ------------------------------------- QFL finalize (int winner -> IoU) -----
__global__ void k_qfl(const float* __restrict__ gtb, const float* __restrict__ pred_d,
                      const float* __restrict__ pts, const float* __restrict__ strides,
                      float* __restrict__ qfl) {
    long i = (long)blockIdx.x * blockDim.x + threadIdx.x;
    if (i >= (long)NB * NC * NA) return;
    int a = (int)(i % NA);
    long bc = i / NA;
    int b = (int)(bc / NC);
    int w = ((const int*)qfl)[i];            // region was int-initialized/scattered
    float out = 0.0f;
    if (w >= 0) {
        const float* gb = gtb + ((size_t)b * NGT + w) * 4;
        float b1[4];
#pragma unroll
        for (int r = 0; r < 4; ++r) b1[r] = fminf(fmaxf(gb[r], 0.0f), IMGF);
        float py = pts[2 * a], px = pts[2 * a + 1], s = strides[a];
        const float* pd = pred_d + (size_t)b * 4 * NA + a;
        float r0 = pd[0] * s;
        float r1 = pd[(size_t)NA] * s;
        float r2 = pd[2 * (size_t)NA] * s;
        float r3 = pd[3 * (size_t)NA] * s;
        float b2[4] = { fminf(fmaxf(py - r0, 0.0f), IMGF),
                        fminf(fmaxf(px - r1, 0.0f), IMGF),
                        fminf(fmaxf(py + r2, 0.0f), IMGF),
                        fminf(fmaxf(px + r3, 0.0f), IMGF) };
        out = iou_f(b1, b2);
    }
    qfl[i] = out;
}

// ---------------------------------------------------------------------------
extern "C" void kernel_launch(void* const* d_in, const int* in_sizes, int n_in,
                              void* d_out, int out_size, void* d_ws, size_t ws_size,
                              hipStream_t stream) {
    (void)in_sizes; (void)n_in; (void)out_size; (void)ws_size;
    const int*   gtc  = (const int*)d_in[0];     // (B,N)
    const float* gtb  = (const float*)d_in[1];   // (B,N,4)
    const float* preg = (const float*)d_in[2];   // (B,4,17,A)

    // workspace layout
    char* ws = (char*)d_ws;
    float* dist2   = (float*)ws;                 ws += sizeof(float) * (size_t)NB * NGT * NA;
    float* pred_d  = (float*)ws;                 ws += sizeof(float) * (size_t)NB * 4 * NA;
    float* pts     = (float*)ws;                 ws += sizeof(float) * (size_t)NA * 2;
    float* boxes   = (float*)ws;                 ws += sizeof(float) * (size_t)NA * 4;
    float* strides = (float*)ws;                 ws += sizeof(float) * (size_t)NA;
    int*   win_d   = (int*)ws;                   ws += sizeof(int)   * (size_t)NB * NA;

    // output layout: qfl | dfl | anchor_pts | strides
    float* qfl   = (float*)d_out;
    float* dfl   = qfl + (size_t)NB * NC * NA;
    float* o_pts = dfl + (size_t)NB * 4 * NA;
    float* o_str = o_pts + (size_t)NA * 2;

    const long nPred = (long)NB * 4 * NA;
    const long nQ    = (long)NB * NC * NA;
    const long nW    = (long)NB * NA;
    const int  tiles = NB * (NGT / 16) * A_TILES;

    k_setup_anchors<<<(NA + 255) / 256, 256, 0, stream>>>(pts, boxes, strides, o_pts, o_str);
    k_pred_d<<<(int)((nPred + 255) / 256), 256, 0, stream>>>(preg, pred_d);
    k_dist2<<<(tiles + 7) / 8, 256, 0, stream>>>(gtb, pts, dist2);
    k_init<<<(int)((nQ + 255) / 256), 256, 0, stream>>>(win_d, (int*)qfl);
    k_assign<<<NB * NGT, 256, 0, stream>>>(dist2, boxes, pts, gtb, gtc, win_d, (int*)qfl);
    k_dfl<<<(int)((nW + 255) / 256), 256, 0, stream>>>(win_d, gtb, pts, strides, dfl);
    k_qfl<<<(int)((nQ + 255) / 256), 256, 0, stream>>>(gtb, pred_d, pts, strides, qfl);
}